// SelfAttention_52819507806728
// MI455X (gfx1250) — compile-verified
//
#include <hip/hip_runtime.h>
#include <hip/hip_bf16.h>
#include <stdint.h>

// ---------------------------------------------------------------------------
// Problem constants (match reference)
// ---------------------------------------------------------------------------
constexpr int kH  = 8;     // heads
constexpr int kD  = 64;    // head dim
constexpr int kDM = 512;   // model dim
constexpr int kB  = 2;     // batch
constexpr int kS  = 2048;  // sequence

typedef __attribute__((ext_vector_type(2))) float v2f;
typedef __attribute__((ext_vector_type(8))) float v8f;

// D = A(16x4) * B(4x16) + C(16x16), fp32 WMMA (CDNA5 / gfx1250)
static __device__ __forceinline__ v8f wmma_f32(v2f a, v2f b, v8f c) {
  return __builtin_amdgcn_wmma_f32_16x16x4_f32(
      /*neg_a=*/false, a, /*neg_b=*/false, b,
      /*c_mod=*/(short)0, c, /*reuse_a=*/false, /*reuse_b=*/false);
}

// Async 16B copy global -> LDS, tracked by ASYNCcnt (CDNA5 §10.8 / §15.18.3).
// %0 = LDS byte address (VGPR), %1 = 64-bit global address (VGPR pair).
static __device__ __forceinline__ void async_ld_b128(uint32_t lds_byte,
                                                     const float* gsrc) {
  asm volatile("global_load_async_to_lds_b128 %0, %1, off"
               :
               : "v"(lds_byte), "v"(gsrc)
               : "memory");
}

static __device__ __forceinline__ void wait_asynccnt0() {
  asm volatile("s_wait_asynccnt 0x0" ::: "memory");
}

// ---------------------------------------------------------------------------
// Kernel 1: per-head QKV projection.
//   x[B,S,H,D] @ W^T + bias  -> out[B,H,S,D]
// Block = 256 threads = 8 waves; wave w handles head w for a 16-row s-tile.
// grid = (B*S/16, 3); blockIdx.y selects Q / K / V.
// ---------------------------------------------------------------------------
__global__ __launch_bounds__(256) void qkv_proj(
    const float* __restrict__ qin, const float* __restrict__ kin,
    const float* __restrict__ vin,
    const float* __restrict__ Wq, const float* __restrict__ bq,
    const float* __restrict__ Wk, const float* __restrict__ bk,
    const float* __restrict__ Wv, const float* __restrict__ bv,
    float* __restrict__ qo, float* __restrict__ ko, float* __restrict__ vo)
{
  const int which = blockIdx.y;
  const float* x    = (which == 0) ? qin : (which == 1) ? kin : vin;
  const float* W    = (which == 0) ? Wq  : (which == 1) ? Wk  : Wv;
  const float* bias = (which == 0) ? bq  : (which == 1) ? bk  : bv;
  float* outp       = (which == 0) ? qo  : (which == 1) ? ko  : vo;

  __shared__ float Wsh[64][68];      // W[e][k], pitch 68 -> conflict-free B reads
  __shared__ float Xs[8][16][68];    // per-wave 16x64 input tile

  const int tid  = threadIdx.x;
  const int wave = tid >> 5;
  const int lane = tid & 31;
  const int h    = wave;
  const int r0   = blockIdx.x * 16;        // flat row (b*S + s)
  const int b    = r0 / kS;
  const int s0   = r0 - b * kS;

  // Cooperative W load: 64x64 floats = 1024 float4
  for (int i = tid; i < 1024; i += 256) {
    const int row = i >> 4, c4 = i & 15;
    const float4 w4 = reinterpret_cast<const float4*>(W)[row * 16 + c4];
    Wsh[row][c4 * 4 + 0] = w4.x; Wsh[row][c4 * 4 + 1] = w4.y;
    Wsh[row][c4 * 4 + 2] = w4.z; Wsh[row][c4 * 4 + 3] = w4.w;
  }
  // Per-wave X tile: rows r0..r0+15, cols h*64..h*64+63
  for (int i = lane; i < 256; i += 32) {
    const int row = i >> 4, c4 = i & 15;
    const float4 x4 =
        reinterpret_cast<const float4*>(x + (size_t)(r0 + row) * kDM + h * 64)[c4];
    Xs[wave][row][c4 * 4 + 0] = x4.x; Xs[wave][row][c4 * 4 + 1] = x4.y;
    Xs[wave][row][c4 * 4 + 2] = x4.z; Xs[wave][row][c4 * 4 + 3] = x4.w;
  }
  __syncthreads();

  const int lm  = lane & 15;
  const int kb2 = (lane >> 4) * 2;   // K offset for this lane half (A & B layout)
  const int hi  = (lane >> 4) * 8;   // C/D row offset for this lane half

  float* op = outp + (size_t)(b * kH + h) * kS * 64;

  for (int nt = 0; nt < 4; ++nt) {
    const float bbv = bias[nt * 16 + lm];
    v8f c;
#pragma unroll
    for (int r = 0; r < 8; ++r) c[r] = bbv;   // bias as C init
#pragma unroll
    for (int kk = 0; kk < 16; ++kk) {         // K = 64 in steps of 4
      v2f a, bm;
      a[0]  = Xs[wave][lm][kk * 4 + kb2];
      a[1]  = Xs[wave][lm][kk * 4 + kb2 + 1];
      bm[0] = Wsh[nt * 16 + lm][kk * 4 + kb2];      // B(k,n) = W[n][k]
      bm[1] = Wsh[nt * 16 + lm][kk * 4 + kb2 + 1];
      c = wmma_f32(a, bm, c);
    }
#pragma unroll
    for (int r = 0; r < 8; ++r)
      op[(size_t)(s0 + r + hi) * 64 + nt * 16 + lm] = c[r];
  }
}

// ---------------------------------------------------------------------------
// Kernel 2: flash attention over K for one (b,h), 8 query tiles per block.
//   scores = Q K^T ; mask==0 -> -1e20 ; softmax(scores * 0.25) ; out = P V
// K/V tiles are double-buffered in LDS and filled with ASYNC loads so tile
// kt+1 streams from memory while tile kt feeds the WMMAs.
// grid = 256 blocks: blockIdx.x = bh*16 + qblk; wave = q-tile within qblk.
// ---------------------------------------------------------------------------
__global__ __launch_bounds__(256) void attn_fwd(
    const float* __restrict__ qp, const float* __restrict__ kp,
    const float* __restrict__ vp, const int* __restrict__ mask,
    float* __restrict__ ctx)
{
  __shared__ float    Qs[8][16][68];    // per-wave Q tile (A operand source)
  __shared__ float    Ks2[2][16][68];   // double-buffered K tile
  __shared__ float    Vs2[2][16][68];   // double-buffered V tile
  __shared__ float    Ps[8][16][20];    // per-wave P tile, C->A layout relay
  __shared__ uint32_t mbits[kS / 32];   // packed mask row (256 B)

  const int tid  = threadIdx.x;
  const int wave = tid >> 5;
  const int lane = tid & 31;
  const int bh   = blockIdx.x >> 4;       // (b*H + h)
  const int qblk = blockIdx.x & 15;
  const int b    = bh >> 3;
  const int q0   = (qblk * 8 + wave) * 16;

  const float* qbase = qp + (size_t)bh * kS * 64;
  const float* kbase = kp + (size_t)bh * kS * 64;
  const float* vbase = vp + (size_t)bh * kS * 64;

  // Pack mask row into a 2048-bit set (once per block).
  if (tid < kS / 32) {
    uint32_t w = 0;
    for (int j = 0; j < 32; ++j)
      w |= (mask[b * kS + tid * 32 + j] != 0 ? 1u : 0u) << j;
    mbits[tid] = w;
  }
  // Per-wave Q tile.
  for (int i = lane; i < 256; i += 32) {
    const int row = i >> 4, c4 = i & 15;
    const float4 t4 =
        reinterpret_cast<const float4*>(qbase + (size_t)(q0 + row) * 64)[c4];
    Qs[wave][row][c4 * 4 + 0] = t4.x; Qs[wave][row][c4 * 4 + 1] = t4.y;
    Qs[wave][row][c4 * 4 + 2] = t4.z; Qs[wave][row][c4 * 4 + 3] = t4.w;
  }

  const int lrow16 = tid >> 4;   // 0..15 (row of the K/V tile this thread fills)
  const int lc4    = tid & 15;   // 0..15 (float4 column)

  // Issue async fill of K/V tile `kt` into buffer `buf` (16 B per thread).
  auto issue_tile = [&](int kt, int buf) {
    const float* kg = kbase + (size_t)(kt * 16 + lrow16) * 64 + lc4 * 4;
    const float* vg = vbase + (size_t)(kt * 16 + lrow16) * 64 + lc4 * 4;
    async_ld_b128((uint32_t)(uintptr_t)&Ks2[buf][lrow16][lc4 * 4], kg);
    async_ld_b128((uint32_t)(uintptr_t)&Vs2[buf][lrow16][lc4 * 4], vg);
  };

  issue_tile(0, 0);   // prologue: tile 0 in flight

  const int lm  = lane & 15;
  const int kb2 = (lane >> 4) * 2;
  const int hi  = (lane >> 4) * 8;

  float mrow[8], lrowst[8];
  v8f acc[4];
#pragma unroll
  for (int r = 0; r < 8; ++r) { mrow[r] = -3.0e38f; lrowst[r] = 0.0f; }
#pragma unroll
  for (int t = 0; t < 4; ++t)
#pragma unroll
    for (int r = 0; r < 8; ++r) acc[t][r] = 0.0f;

  constexpr int NT = kS / 16;   // 128 key tiles
  for (int kt = 0; kt < NT; ++kt) {
    const int cur = kt & 1;
    wait_asynccnt0();   // this wave's fills for tile kt (and older) have landed
    __syncthreads();    // all waves' fills landed; prev iter fully consumed
    if (kt + 1 < NT) issue_tile(kt + 1, cur ^ 1);  // overlap with compute below

    // scores S = Q K^T  (16x16 tile, 16 WMMA K=4 steps)
    v8f sc;
#pragma unroll
    for (int r = 0; r < 8; ++r) sc[r] = 0.0f;
#pragma unroll
    for (int kk = 0; kk < 16; ++kk) {
      v2f a, bm;
      a[0]  = Qs[wave][lm][kk * 4 + kb2];
      a[1]  = Qs[wave][lm][kk * 4 + kb2 + 1];
      bm[0] = Ks2[cur][lm][kk * 4 + kb2];       // B(d,n) = K[n][d]
      bm[1] = Ks2[cur][lm][kk * 4 + kb2 + 1];
      sc = wmma_f32(a, bm, sc);
    }

    // mask then scale (ref: softmax((-1e20 | s) * 2/H), 2/H == 0.25)
    const int col = kt * 16 + lm;               // key column for this lane
    const int mv  = (mbits[col >> 5] >> (col & 31)) & 1;
#pragma unroll
    for (int r = 0; r < 8; ++r) sc[r] = mv ? sc[r] * 0.25f : -2.5e19f;

    // online softmax: per-row (16-lane group) max/sum reductions
#pragma unroll
    for (int r = 0; r < 8; ++r) {
      float v = sc[r];
      v = fmaxf(v, __shfl_xor(v, 1, 32));
      v = fmaxf(v, __shfl_xor(v, 2, 32));
      v = fmaxf(v, __shfl_xor(v, 4, 32));
      v = fmaxf(v, __shfl_xor(v, 8, 32));
      const float nm   = fmaxf(mrow[r], v);
      const float corr = __expf(mrow[r] - nm);
      mrow[r] = nm;
      const float p = __expf(sc[r] - nm);
      Ps[wave][r + hi][lm] = p;          // C-layout -> LDS
      float s = p;
      s += __shfl_xor(s, 1, 32);
      s += __shfl_xor(s, 2, 32);
      s += __shfl_xor(s, 4, 32);
      s += __shfl_xor(s, 8, 32);
      lrowst[r] = lrowst[r] * corr + s;
#pragma unroll
      for (int t = 0; t < 4; ++t) acc[t][r] *= corr;
    }
    __syncthreads();  // Ps visible for A-layout reads

    // acc += P * V   (4 WMMA K=4 steps per 16-col d-tile)
#pragma unroll
    for (int kk = 0; kk < 4; ++kk) {
      v2f a;
      a[0] = Ps[wave][lm][kk * 4 + kb2];
      a[1] = Ps[wave][lm][kk * 4 + kb2 + 1];
#pragma unroll
      for (int t = 0; t < 4; ++t) {
        v2f bm;
        bm[0] = Vs2[cur][kk * 4 + kb2][t * 16 + lm];     // B(n,d) = V[n][d]
        bm[1] = Vs2[cur][kk * 4 + kb2 + 1][t * 16 + lm];
        acc[t] = wmma_f32(a, bm, acc[t]);
      }
    }
  }

  float* cb = ctx + (size_t)bh * kS * 64;
#pragma unroll
  for (int t = 0; t < 4; ++t)
#pragma unroll
    for (int r = 0; r < 8; ++r)
      cb[(size_t)(q0 + r + hi) * 64 + t * 16 + lm] = acc[t][r] / lrowst[r];
}

// ---------------------------------------------------------------------------
// Kernel 3: output projection with folded (d,h) permutation.
//   out[r][n] = sum_{h,d} ctx[b,h,s,d] * Wo[n][d*8+h] + bo[n]
// Block = 8 waves = 128 rows x 64 cols; K chunk hc == head hc, so the
// permuted Wo column index is simply d*8 + hc.
// grid = (B*S/128, DM/64)
// ---------------------------------------------------------------------------
__global__ __launch_bounds__(256) void out_proj(
    const float* __restrict__ ctx, const float* __restrict__ Wo,
    const float* __restrict__ bo, float* __restrict__ out)
{
  __shared__ float Wc[64][68];       // Wo chunk [n][d], shared by all 8 waves
  __shared__ float Xs[8][16][68];    // per-wave ctx chunk

  const int tid  = threadIdx.x;
  const int wave = tid >> 5;
  const int lane = tid & 31;
  const int r0   = blockIdx.x * 128 + wave * 16;
  const int n0   = blockIdx.y * 64;
  const int lm   = lane & 15;
  const int kb2  = (lane >> 4) * 2;
  const int hi   = (lane >> 4) * 8;

  v8f acc[4];
#pragma unroll
  for (int t = 0; t < 4; ++t) {
    const float bbv = bo[n0 + t * 16 + lm];
#pragma unroll
    for (int r = 0; r < 8; ++r) acc[t][r] = bbv;
  }

  for (int hc = 0; hc < kH; ++hc) {
    __syncthreads();
    // Wc[n][d] = Wo[n0+n][d*8 + hc]  (permutation folded in)
    for (int i = tid; i < 4096; i += 256) {
      const int n = i >> 6, d = i & 63;
      Wc[n][d] = Wo[(size_t)(n0 + n) * kDM + d * kH + hc];
    }
    // Xs: rows r0..r0+15 of ctx[b, hc, s, :]
    for (int i = lane; i < 256; i += 32) {
      const int row = i >> 4, c4 = i & 15;
      const int r = r0 + row, b = r >> 11, s = r & (kS - 1);
      const float4 t4 = reinterpret_cast<const float4*>(
          ctx + ((size_t)(b * kH + hc) * kS + s) * 64)[c4];
      Xs[wave][row][c4 * 4 + 0] = t4.x; Xs[wave][row][c4 * 4 + 1] = t4.y;
      Xs[wave][row][c4 * 4 + 2] = t4.z; Xs[wave][row][c4 * 4 + 3] = t4.w;
    }
    __syncthreads();

#pragma unroll
    for (int kk = 0; kk < 16; ++kk) {
      v2f a;
      a[0] = Xs[wave][lm][kk * 4 + kb2];
      a[1] = Xs[wave][lm][kk * 4 + kb2 + 1];
#pragma unroll
      for (int t = 0; t < 4; ++t) {
        v2f bm;
        bm[0] = Wc[t * 16 + lm][kk * 4 + kb2];
        bm[1] = Wc[t * 16 + lm][kk * 4 + kb2 + 1];
        acc[t] = wmma_f32(a, bm, acc[t]);
      }
    }
  }

#pragma unroll
  for (int t = 0; t < 4; ++t)
#pragma unroll
    for (int r = 0; r < 8; ++r)
      out[(size_t)(r0 + r + hi) * kDM + n0 + t * 16 + lm] = acc[t][r];
}

// ---------------------------------------------------------------------------
// Launch
// ---------------------------------------------------------------------------
extern "C" void kernel_launch(void* const* d_in, const int* in_sizes, int n_in,
                              void* d_out, int out_size, void* d_ws, size_t ws_size,
                              hipStream_t stream) {
  const float* query = (const float*)d_in[0];
  const float* key   = (const float*)d_in[1];
  const float* value = (const float*)d_in[2];
  const float* Wq    = (const float*)d_in[3];
  const float* bq    = (const float*)d_in[4];
  const float* Wk    = (const float*)d_in[5];
  const float* bk    = (const float*)d_in[6];
  const float* Wv    = (const float*)d_in[7];
  const float* bv    = (const float*)d_in[8];
  const float* Wo    = (const float*)d_in[9];
  const float* bo    = (const float*)d_in[10];
  const int*   mask  = (const int*)d_in[11];
  float* outp = (float*)d_out;

  const size_t per = (size_t)kB * kH * kS * kD;  // 2M floats = 8 MB
  float* q_ws = (float*)d_ws;
  float* k_ws = q_ws + per;
  float* v_ws = k_ws + per;
  float* c_ws = v_ws + per;

  qkv_proj<<<dim3(kB * kS / 16, 3), 256, 0, stream>>>(
      query, key, value, Wq, bq, Wk, bk, Wv, bv, q_ws, k_ws, v_ws);

  attn_fwd<<<dim3(kB * kH * (kS / 16) / 8), 256, 0, stream>>>(
      q_ws, k_ws, v_ws, mask, c_ws);

  out_proj<<<dim3(kB * kS / 128, kDM / 64), 256, 0, stream>>>(
      c_ws, Wo, bo, outp);
}